// Lego_Message_Passing_Mnist_With_Msg_Fc_50689204027500
// MI455X (gfx1250) — compile-verified
//
#include <hip/hip_runtime.h>

typedef __attribute__((ext_vector_type(2))) float v2f;
typedef __attribute__((ext_vector_type(8))) float v8f;

namespace {
constexpr int Bn = 32;     // batch
constexpr int Nn = 128;    // nodes
constexpr int Dn = 4;      // node feature dim
constexpr int De = 4;      // edge feature dim
constexpr int Hn = 128;    // hidden
constexpr int SW = 132;    // padded LDS row stride (floats): 132%64=4 -> conflict-free K-major reads
}

// One workgroup (8 waves of 32) per (b, i).
// Wave w = (m, g): m = w>>2 owns j-tiles {4m..4m+3} (64 rows), g = w&3 owns
// n-tiles {2g, 2g+1} (32 h-columns). 4x2 register blocking so W fragments
// amortize over 64 M-rows (4 global loads + 4 ds loads per 8 WMMA).
__global__ __launch_bounds__(256, 1) void mpnn_fused(
    const float* __restrict__ node,   // [B,N,4]
    const float* __restrict__ adj,    // [B,N,N]
    const float* __restrict__ tgt,    // [B,4]
    const float* __restrict__ edgef,  // [B,N,N,4]
    const float* __restrict__ Wed,    // [12,128]
    const float* __restrict__ bed,    // [128]
    const float* __restrict__ Wms,    // [136,128]
    const float* __restrict__ bms,    // [128]
    const float* __restrict__ Wft,    // [132,128]
    const float* __restrict__ bft,    // [128]
    float* __restrict__ out_end,      // [B,N,128]
    float* __restrict__ out_edge,     // [B,N,N,128]
    float* __restrict__ out_tgt)      // [B*4]
{
    __shared__ float strip[Nn * SW];       // full 128x128 edge_h tile (padded rows), WG-shared
    float* red  = strip;                   // aliased after the strip is consumed: [2][128]
    float* msum = strip + 2 * Hn;          // [128]

    const int tid  = threadIdx.x;
    const int lane = tid & 31;
    const int wave = tid >> 5;
    const int l16  = lane & 15;
    const int lh   = lane >> 4;            // lane half (wave32): 0 or 1

    const int g = wave & 3;                // n-tile group: nt = 2g + t, t in {0,1}
    const int m = wave >> 2;               // j-tile group: Mt = 4m + mt, mt in {0..3}

    const int b = blockIdx.x >> 7;
    const int i = blockIdx.x & (Nn - 1);

    // ---- per-lane A-fragment source data ------------------------------------
    float nI[Dn];                          // node_i, broadcast to all lanes
    #pragma unroll
    for (int k = 0; k < Dn; ++k) nI[k] = node[(b * Nn + i) * Dn + k];

    // A fragment layout (f32 16x4): lane = M (l16), K = kt*4 + 2*lh + v
    v2f aI;                                // K 0..3 : node_i (same for every M row)
    aI[0] = nI[2 * lh]; aI[1] = nI[2 * lh + 1];

    v2f aJ[4], aE[4];                      // K 4..7 : node_j, K 8..11 : edge feature
    #pragma unroll
    for (int mt = 0; mt < 4; ++mt) {
        const int jrow = (4 * m + mt) * 16 + l16;
        float nJ0 = node[(b * Nn + jrow) * Dn + 2 * lh];
        float nJ1 = node[(b * Nn + jrow) * Dn + 2 * lh + 1];
        aJ[mt][0] = nJ0; aJ[mt][1] = nJ1;
        const size_t eb = (((size_t)(b * Nn + i)) * Nn + jrow) * De;
        aE[mt][0] = edgef[eb + 2 * lh];
        aE[mt][1] = edgef[eb + 2 * lh + 1];
    }

    // ---- GEMM 1: edge_h = relu([nI|nJ|eF](64x12) @ W_edge(12x128) + b) ------
    v8f acc[4][2];
    #pragma unroll
    for (int mt = 0; mt < 4; ++mt)
        #pragma unroll
        for (int t = 0; t < 2; ++t) acc[mt][t] = v8f{};

    #pragma unroll
    for (int kt = 0; kt < 3; ++kt) {                      // K=12 -> 3 k-steps
        const int Kb = kt * 4 + 2 * lh;
        v2f bf[2];                                        // B layout: lane=N, (VGPR,half)=K
        #pragma unroll
        for (int t = 0; t < 2; ++t) {
            const int nc = (2 * g + t) * 16 + l16;
            bf[t][0] = Wed[Kb * Hn + nc];
            bf[t][1] = Wed[(Kb + 1) * Hn + nc];
        }
        #pragma unroll
        for (int mt = 0; mt < 4; ++mt) {
            v2f a = (kt == 0) ? aI : ((kt == 1) ? aJ[mt] : aE[mt]);
            #pragma unroll
            for (int t = 0; t < 2; ++t)
                acc[mt][t] = __builtin_amdgcn_wmma_f32_16x16x4_f32(
                    false, a, false, bf[t], (short)0, acc[mt][t], false, false);
        }
    }

    // bias + relu; store edge_h to global (output #2) and the shared LDS strip
    #pragma unroll
    for (int t = 0; t < 2; ++t) {
        const int h  = (2 * g + t) * 16 + l16;
        const float be = bed[h];
        #pragma unroll
        for (int mt = 0; mt < 4; ++mt) {
            #pragma unroll
            for (int r = 0; r < 8; ++r) {                 // C layout: M = r + 8*lh, N = l16
                float v = acc[mt][t][r] + be;
                v = v > 0.f ? v : 0.f;
                const int j = (4 * m + mt) * 16 + r + 8 * lh;
                strip[j * SW + h] = v;
                out_edge[(((size_t)(b * Nn + i)) * Nn + j) * Hn + h] = v;
            }
        }
    }
    __syncthreads();   // strip is written by all waves, read across waves below

    // ---- GEMM 2: msg = relu([node_i|node_j|edge_h](64x136) @ W_msg + b) -----
    v8f mac[4][2];
    #pragma unroll
    for (int mt = 0; mt < 4; ++mt)
        #pragma unroll
        for (int t = 0; t < 2; ++t) mac[mt][t] = v8f{};

    #pragma unroll
    for (int kt = 0; kt < 2; ++kt) {                      // K 0..7 : node_node
        const int Kb = kt * 4 + 2 * lh;
        v2f bf[2];
        #pragma unroll
        for (int t = 0; t < 2; ++t) {
            const int nc = (2 * g + t) * 16 + l16;
            bf[t][0] = Wms[Kb * Hn + nc];
            bf[t][1] = Wms[(Kb + 1) * Hn + nc];
        }
        #pragma unroll
        for (int mt = 0; mt < 4; ++mt) {
            v2f a = (kt == 0) ? aI : aJ[mt];
            #pragma unroll
            for (int t = 0; t < 2; ++t)
                mac[mt][t] = __builtin_amdgcn_wmma_f32_16x16x4_f32(
                    false, a, false, bf[t], (short)0, mac[mt][t], false, false);
        }
    }
    for (int kt = 2; kt < 34; ++kt) {                     // K 8..135 : edge_h from LDS
        const int Kb = kt * 4 + 2 * lh;
        v2f bf[2];
        #pragma unroll
        for (int t = 0; t < 2; ++t) {
            const int nc = (2 * g + t) * 16 + l16;
            bf[t][0] = Wms[Kb * Hn + nc];
            bf[t][1] = Wms[(Kb + 1) * Hn + nc];
        }
        #pragma unroll
        for (int mt = 0; mt < 4; ++mt) {
            const float* ap = &strip[((4 * m + mt) * 16 + l16) * SW + (Kb - 8)];
            v2f a; a[0] = ap[0]; a[1] = ap[1];            // consecutive -> ds_load_b64
            #pragma unroll
            for (int t = 0; t < 2; ++t)
                mac[mt][t] = __builtin_amdgcn_wmma_f32_16x16x4_f32(
                    false, a, false, bf[t], (short)0, mac[mt][t], false, false);
        }
    }

    // ---- bias+relu, adjacency weighting, reduce over this wave's 64 j's -----
    float aj[4][8];
    #pragma unroll
    for (int mt = 0; mt < 4; ++mt)
        #pragma unroll
        for (int r = 0; r < 8; ++r)
            aj[mt][r] = adj[((size_t)(b * Nn + i)) * Nn + (4 * m + mt) * 16 + r + 8 * lh];

    float ps[2];
    #pragma unroll
    for (int t = 0; t < 2; ++t) {
        const float bm = bms[(2 * g + t) * 16 + l16];
        float s = 0.f;
        #pragma unroll
        for (int mt = 0; mt < 4; ++mt)
            #pragma unroll
            for (int r = 0; r < 8; ++r) {
                float v = mac[mt][t][r] + bm;
                v = v > 0.f ? v : 0.f;
                s += v * aj[mt][r];                       // weight by adjacency, sum over j
            }
        s += __shfl_xor(s, 16, 32);                       // fold the two lane halves (wave32)
        ps[t] = s;
    }

    __syncthreads();          // all waves done reading strip; alias LDS as red/msum

    if (lh == 0) {
        #pragma unroll
        for (int t = 0; t < 2; ++t)
            red[m * Hn + (2 * g + t) * 16 + l16] = ps[t]; // [m][h]
    }
    __syncthreads();

    if (tid < Hn)                                         // final j reduction: 2 row-groups
        msum[tid] = red[tid] + red[Hn + tid];             // message[b,i,tid]
    __syncthreads();

    // ---- final FC: ending = relu([message | node_i] @ W_feat + b_feat) ------
    if (tid < Hn) {
        float o = bft[tid];
        for (int k = 0; k < Hn; ++k) o += msum[k] * Wft[k * Hn + tid];
        #pragma unroll
        for (int k = 0; k < Dn; ++k) o += nI[k] * Wft[(Hn + k) * Hn + tid];
        o = o > 0.f ? o : 0.f;
        out_end[((size_t)(b * Nn + i)) * Hn + tid] = o;
    }

    // pass-through output #3
    if (blockIdx.x == 0 && tid < Bn * 4) out_tgt[tid] = tgt[tid];
}

extern "C" void kernel_launch(void* const* d_in, const int* in_sizes, int n_in,
                              void* d_out, int out_size, void* d_ws, size_t ws_size,
                              hipStream_t stream) {
    const float* node = (const float*)d_in[0];   // node_feature
    const float* adj  = (const float*)d_in[1];   // adjacency
    const float* tgt  = (const float*)d_in[2];   // target_information
    const float* edgf = (const float*)d_in[3];   // edge_feature
    //                  d_in[4] = node_mask (unused by forward)
    const float* Wed  = (const float*)d_in[5];
    const float* bed  = (const float*)d_in[6];
    const float* Wms  = (const float*)d_in[7];
    const float* bms  = (const float*)d_in[8];
    const float* Wft  = (const float*)d_in[9];
    const float* bft  = (const float*)d_in[10];

    float* out      = (float*)d_out;
    float* out_end  = out;                                            // [32,128,128]
    float* out_edge = out + (size_t)Bn * Nn * Hn;                     // [32,128,128,128]
    float* out_tgt  = out + (size_t)Bn * Nn * Hn + (size_t)Bn * Nn * Nn * Hn;

    dim3 grid(Bn * Nn);   // 4096 workgroups: one per (b, i)
    dim3 block(256);      // 8 waves (wave32)
    mpnn_fused<<<grid, block, 0, stream>>>(node, adj, tgt, edgf,
                                           Wed, bed, Wms, bms, Wft, bft,
                                           out_end, out_edge, out_tgt);
}